// CosformerAttention_1451698946423
// MI455X (gfx1250) — compile-verified
//
#include <hip/hip_runtime.h>
#include <hip/hip_bf16.h>

// Problem constants (reference: N=2, H=8, T=1024, HID=64, EMBED=512)
#define T_LEN   1024
#define HID     64
#define NH      16          // N*H
#define CS      64          // chunk size along T
#define NCHUNK  (T_LEN / CS)
#define EMBED   512
#define STATE_D 128         // feature dim after sin||cos concat

typedef __attribute__((ext_vector_type(2))) float v2f;
typedef __attribute__((ext_vector_type(8))) float v8f;

// V_WMMA_F32_16X16X4_F32: D(16x16 f32) = A(16x4 f32) * B(4x16 f32) + C
__device__ __forceinline__ v8f wmma_f32_4(v2f a, v2f b, v8f c) {
  return __builtin_amdgcn_wmma_f32_16x16x4_f32(false, a, false, b, (short)0, c,
                                               false, false);
}

// LDS row strides (floats), padded so column-style fragment loads are
// bank-conflict-free (stride mod 64 == 4 -> 16 lanes hit 16 distinct banks).
#define SQ_S 132   // Q_/K_ : 64 rows x 128 cols (sin||cos)
#define SV_S 68    // V     : 64 x 64
#define SS_S 68    // prefix state : 128 x 64
#define SA_S 68    // scores/A-tiles : 64 x 64

// Load one (relu + sin/cos reweighted) Q or K chunk into LDS (V is loaded by the caller).
__device__ __forceinline__ void load_chunk_qk(
    const float* __restrict__ src, float* __restrict__ dst /*CS*SQ_S*/,
    int t0, int tid) {
  for (int i = tid; i < CS * HID; i += 256) {
    const int r = i >> 6;
    const int d = i & 63;
    const int t = t0 + r;
    const float ang = 1.57079632679489662f * (float)(t + 1) *
                      (1.0f / (float)T_LEN);
    float x = src[(size_t)t * HID + d];
    x = x > 0.0f ? x : 0.0f;
    dst[r * SQ_S + d]      = x * __sinf(ang);
    dst[r * SQ_S + d + 64] = x * __cosf(ang);
  }
}

// ---------------------------------------------------------------------------
// Pass 1: per (head, chunk) partial state  S_c = K_^T @ V  (128x64, K=64)
//         and z_c = colsum(K_).  Fully parallel: 16x16 = 256 workgroups.
// ---------------------------------------------------------------------------
__global__ __launch_bounds__(256) void chunk_state_kernel(
    const float* __restrict__ k, const float* __restrict__ v,
    float* __restrict__ states, float* __restrict__ zbuf)
{
  __shared__ float sK[CS * SQ_S];
  __shared__ float sV[CS * SV_S];
  const int chunk = blockIdx.x;       // 0..15
  const int nh    = blockIdx.y;       // 0..15
  const int tid   = threadIdx.x;
  const int lane  = tid & 31;
  const int wave  = tid >> 5;
  const int ln15  = lane & 15;
  const int lhi   = lane >> 4;
  const int t0    = chunk * CS;

  load_chunk_qk(k + (size_t)nh * T_LEN * HID, sK, t0, tid);
  for (int i = tid; i < CS * HID; i += 256) {
    const int r = i >> 6, d = i & 63;
    sV[r * SV_S + d] = v[((size_t)(t0 + r) * NH + nh) * HID + d];
  }
  __syncthreads();

  const size_t sbase = (size_t)(nh * NCHUNK + chunk) * STATE_D * HID;
  for (int tile = wave; tile < 32; tile += 8) {
    const int m0 = (tile >> 2) * 16;       // d rows, 0..127
    const int n0 = (tile & 3) * 16;        // v cols
    const int dcol = m0 + ln15;
    const int bn   = n0 + ln15;
    const int mb   = m0 + (lhi << 3);
    v8f acc = {};
    for (int kk = 0; kk < 64; kk += 4) {
      const int kr = kk + (lhi << 1);
      v2f a, b;
      a.x = sK[kr * SQ_S + dcol];          // (K_^T)[d][row] = K_[row][d]
      a.y = sK[(kr + 1) * SQ_S + dcol];
      b.x = sV[kr * SV_S + bn];
      b.y = sV[(kr + 1) * SV_S + bn];
      acc = wmma_f32_4(a, b, acc);
    }
    for (int r2 = 0; r2 < 8; ++r2)
      states[sbase + (size_t)(mb + r2) * HID + bn] = acc[r2];
  }
  if (tid < STATE_D) {
    float s = 0.0f;
    for (int r = 0; r < CS; ++r) s += sK[r * SQ_S + tid];
    zbuf[(size_t)(nh * NCHUNK + chunk) * STATE_D + tid] = s;
  }
}

// ---------------------------------------------------------------------------
// Pass 2: in-place EXCLUSIVE prefix sum over chunks for states and z.
// One thread per (head, d, m) element; coalesced along m.
// ---------------------------------------------------------------------------
__global__ void prefix_scan_kernel(float* __restrict__ states,
                                   float* __restrict__ zbuf)
{
  const int gid = blockIdx.x * blockDim.x + threadIdx.x;
  const int per_head = STATE_D * HID;            // 8192
  if (gid < NH * per_head) {
    const int head = gid / per_head;
    const int rem  = gid % per_head;
    const size_t base = (size_t)head * NCHUNK * per_head + rem;
    float run = 0.0f;
    for (int c = 0; c < NCHUNK; ++c) {
      const size_t idx = base + (size_t)c * per_head;
      const float t = states[idx];
      states[idx] = run;
      run += t;
    }
  }
  if (gid < NH * STATE_D) {
    const int head = gid >> 7;
    const int d    = gid & (STATE_D - 1);
    const size_t base = (size_t)head * NCHUNK * STATE_D + d;
    float run = 0.0f;
    for (int c = 0; c < NCHUNK; ++c) {
      const size_t idx = base + (size_t)c * STATE_D;
      const float t = zbuf[idx];
      zbuf[idx] = run;
      run += t;
    }
  }
}

// ---------------------------------------------------------------------------
// Pass 3: per (head, chunk) output.  scores = Q_@K_^T (masked), then
// numer = scores@V + Q_@P_c (exclusive prefix state), denom = rowsum + Q_.z.
// Fully parallel: 256 workgroups.
// ---------------------------------------------------------------------------
__global__ __launch_bounds__(256) void chunk_output_kernel(
    const float* __restrict__ q, const float* __restrict__ k,
    const float* __restrict__ v, const float* __restrict__ states,
    const float* __restrict__ zbuf, float* __restrict__ attn_ws)
{
  __shared__ float sQ[CS * SQ_S];
  __shared__ float sK[CS * SQ_S];
  __shared__ float sV[CS * SV_S];
  __shared__ float sP[STATE_D * SS_S];   // exclusive prefix state
  __shared__ float sA[CS * SA_S];        // masked scores
  __shared__ float sz[STATE_D];
  __shared__ float sden[CS];

  const int chunk = blockIdx.x;
  const int nh    = blockIdx.y;
  const int tid   = threadIdx.x;
  const int lane  = tid & 31;
  const int wave  = tid >> 5;
  const int ln15  = lane & 15;
  const int lhi   = lane >> 4;
  const int t0    = chunk * CS;
  const int nb    = nh >> 3;             // batch index
  const int hh    = nh & 7;              // head index

  load_chunk_qk(q + (size_t)nh * T_LEN * HID, sQ, t0, tid);
  load_chunk_qk(k + (size_t)nh * T_LEN * HID, sK, t0, tid);
  for (int i = tid; i < CS * HID; i += 256) {
    const int r = i >> 6, d = i & 63;
    sV[r * SV_S + d] = v[((size_t)(t0 + r) * NH + nh) * HID + d];
  }
  {
    const size_t sbase = (size_t)(nh * NCHUNK + chunk) * STATE_D * HID;
    for (int i = tid; i < STATE_D * HID; i += 256) {
      const int d = i >> 6, m = i & 63;
      sP[d * SS_S + m] = states[sbase + i];
    }
    if (tid < STATE_D)
      sz[tid] = zbuf[(size_t)(nh * NCHUNK + chunk) * STATE_D + tid];
  }
  __syncthreads();

  // ---- Phase A: scores = Q_ @ K_^T (64x64, K=128), causal mask n<=m ----
  for (int tile = wave; tile < 16; tile += 8) {
    const int m0 = (tile >> 2) * 16;
    const int n0 = (tile & 3) * 16;
    const int am = m0 + ln15;
    const int bn = n0 + ln15;
    v8f acc = {};
    for (int kk = 0; kk < STATE_D; kk += 4) {
      const int kr = kk + (lhi << 1);
      v2f a, b;
      a.x = sQ[am * SQ_S + kr];
      a.y = sQ[am * SQ_S + kr + 1];
      b.x = sK[bn * SQ_S + kr];            // (K_^T)[kr][bn] = K_[bn][kr]
      b.y = sK[bn * SQ_S + kr + 1];
      acc = wmma_f32_4(a, b, acc);
    }
    const int mb = m0 + (lhi << 3);
    for (int r2 = 0; r2 < 8; ++r2) {
      const int m = mb + r2;
      sA[m * SA_S + bn] = (bn <= m) ? acc[r2] : 0.0f;
    }
  }
  __syncthreads();

  // ---- denominator: rowsum(masked scores) + Q_ . z_prev ----
  if (tid < CS) {
    const int m = tid;
    float s = 0.0f;
    for (int n2 = 0; n2 <= m; ++n2) s += sA[m * SA_S + n2];
    for (int d2 = 0; d2 < STATE_D; ++d2) s += sQ[m * SQ_S + d2] * sz[d2];
    sden[m] = fmaxf(s, 1e-6f);
  }
  __syncthreads();

  // ---- Phase B: numer = scores @ V + Q_ @ P ; divide ; stage to d_ws ----
  for (int tile = wave; tile < 16; tile += 8) {
    const int m0 = (tile >> 2) * 16;
    const int n0 = (tile & 3) * 16;
    const int am = m0 + ln15;
    const int bn = n0 + ln15;
    v8f acc = {};
    for (int kk = 0; kk < 64; kk += 4) {             // intra-chunk part
      const int kr = kk + (lhi << 1);
      v2f a, b;
      a.x = sA[am * SA_S + kr];
      a.y = sA[am * SA_S + kr + 1];
      b.x = sV[kr * SV_S + bn];
      b.y = sV[(kr + 1) * SV_S + bn];
      acc = wmma_f32_4(a, b, acc);
    }
    for (int kk = 0; kk < STATE_D; kk += 4) {        // inter-chunk (prefix)
      const int kr = kk + (lhi << 1);
      v2f a, b;
      a.x = sQ[am * SQ_S + kr];
      a.y = sQ[am * SQ_S + kr + 1];
      b.x = sP[kr * SS_S + bn];
      b.y = sP[(kr + 1) * SS_S + bn];
      acc = wmma_f32_4(a, b, acc);
    }
    const int mb = m0 + (lhi << 3);
    for (int r2 = 0; r2 < 8; ++r2) {
      const int m = mb + r2;
      const int t = t0 + m;
      const float val = acc[r2] / sden[m];
      // staged as (T, N, H*HID) row-major for the out-proj GEMM
      attn_ws[((size_t)t * 2 + nb) * EMBED + hh * 64 + bn] = val;
    }
  }
}

// ---------------------------------------------------------------------------
// Out-proj: out = attn @ W^T + b, fused with (T,N,512)->(N,H,T,HID) scatter.
// GEMM M=2048, N=512, K=512; 64x64 tiles per workgroup, 256 workgroups.
// ---------------------------------------------------------------------------
__global__ __launch_bounds__(256) void outproj_kernel(
    const float* __restrict__ attn, const float* __restrict__ W,
    const float* __restrict__ bvec, float* __restrict__ ctx)
{
  __shared__ float sAt[64 * SA_S];
  __shared__ float sW[64 * SA_S];
  const int tid  = threadIdx.x;
  const int lane = tid & 31;
  const int wave = tid >> 5;
  const int ln15 = lane & 15;
  const int lhi  = lane >> 4;
  const int row0 = blockIdx.x * 64;   // rows = t*2 + n  (2048 total)
  const int e0   = blockIdx.y * 64;   // output features (512 total)

  v8f acc0 = {}, acc1 = {};
  for (int f0 = 0; f0 < EMBED; f0 += 64) {
    for (int i = tid; i < 64 * 64; i += 256) {
      const int r = i >> 6, cc = i & 63;
      sAt[r * SA_S + cc] = attn[(size_t)(row0 + r) * EMBED + f0 + cc];
      sW [r * SA_S + cc] = W   [(size_t)(e0  + r) * EMBED + f0 + cc];
    }
    __syncthreads();
    for (int half = 0; half < 2; ++half) {
      const int tile = wave + half * 8;
      const int m0 = (tile >> 2) * 16;
      const int n0 = (tile & 3) * 16;
      const int am = m0 + ln15;
      const int bn = n0 + ln15;
      v8f acc = half ? acc1 : acc0;
      for (int kk = 0; kk < 64; kk += 4) {
        const int kr = kk + (lhi << 1);
        v2f a, b;
        a.x = sAt[am * SA_S + kr];
        a.y = sAt[am * SA_S + kr + 1];
        b.x = sW[bn * SA_S + kr];          // (W^T)[kr][e] = W[e][kr]
        b.y = sW[bn * SA_S + kr + 1];
        acc = wmma_f32_4(a, b, acc);
      }
      if (half) acc1 = acc; else acc0 = acc;
    }
    __syncthreads();
  }

  for (int half = 0; half < 2; ++half) {
    const v8f acc = half ? acc1 : acc0;
    const int tile = wave + half * 8;
    const int m0 = (tile >> 2) * 16;
    const int n0 = (tile & 3) * 16;
    const int e  = e0 + n0 + ln15;
    const int mb = m0 + (lhi << 3);
    const float bias = bvec[e];
    const int hh = e >> 6, d = e & 63;
    for (int r2 = 0; r2 < 8; ++r2) {
      const int row = row0 + mb + r2;
      const int t = row >> 1, nb = row & 1;
      ctx[(((size_t)nb * 8 + hh) * T_LEN + t) * HID + d] = acc[r2] + bias;
    }
  }
}

// ---------------------------------------------------------------------------
// Zero the (N,H,T,T) probs output (64 MB) — pure store bandwidth (~3 us).
// ---------------------------------------------------------------------------
__global__ void zero_probs_kernel(float4* __restrict__ p, int n4) {
  const int i = blockIdx.x * blockDim.x + threadIdx.x;
  if (i < n4) p[i] = make_float4(0.0f, 0.0f, 0.0f, 0.0f);
}

extern "C" void kernel_launch(void* const* d_in, const int* in_sizes, int n_in,
                              void* d_out, int out_size, void* d_ws, size_t ws_size,
                              hipStream_t stream) {
  (void)in_sizes; (void)n_in; (void)out_size; (void)ws_size;
  const float* q    = (const float*)d_in[0];
  const float* k    = (const float*)d_in[1];
  const float* v    = (const float*)d_in[2];
  const float* W    = (const float*)d_in[3];
  const float* bout = (const float*)d_in[4];

  float* ctx   = (float*)d_out;                          // 2*8*1024*64
  float* probs = ctx + (size_t)2 * 8 * 1024 * 64;        // 2*8*1024*1024 zeros

  // Workspace layout (floats):
  //   [0, 1048576)            attn   (T*N, 512)
  //   [1048576, 3145728)      states (NH, NCHUNK, 128, 64) -> excl. prefix
  //   [3145728, 3178496)      z      (NH, NCHUNK, 128)     -> excl. prefix
  float* attn_ws = (float*)d_ws;
  float* states  = attn_ws + (size_t)2048 * EMBED;
  float* zbuf    = states + (size_t)NH * NCHUNK * STATE_D * HID;

  // probs zero-fill (independent, pure bandwidth)
  const int n4 = (2 * 8 * 1024 * 1024) / 4;
  zero_probs_kernel<<<(n4 + 255) / 256, 256, 0, stream>>>((float4*)probs, n4);

  dim3 gchunks(NCHUNK, NH);   // 16 x 16 = 256 workgroups

  // Pass 1: per-chunk states (parallel)
  chunk_state_kernel<<<gchunks, 256, 0, stream>>>(k, v, states, zbuf);

  // Pass 2: exclusive prefix over chunks (flattens the serial scan)
  const int nscan = NH * STATE_D * HID;   // 131072 threads
  prefix_scan_kernel<<<(nscan + 255) / 256, 256, 0, stream>>>(states, zbuf);

  // Pass 3: per-chunk outputs (parallel)
  chunk_output_kernel<<<gchunks, 256, 0, stream>>>(q, k, v, states, zbuf,
                                                   attn_ws);

  // Out-proj + bias + transpose to (N,H,T,HID)
  dim3 gproj(2048 / 64, EMBED / 64);
  outproj_kernel<<<gproj, 256, 0, stream>>>(attn_ws, W, bout, ctx);
}